// NormDropout_3831110828066
// MI455X (gfx1250) — compile-verified
//
#include <hip/hip_runtime.h>
#include <hip/hip_bf16.h>
#include <stdint.h>

typedef __attribute__((ext_vector_type(2))) float v2f;
typedef __attribute__((ext_vector_type(8))) float v8f;

#define C_CH   256
#define H_IMG  512
#define W_IMG  512
#define FW     256
#define LCOLS  65536
#define KEEP   20
#define T2COLS (LCOLS - KEEP)          // 65516
#define OUTC   (KEEP * 2 + T2COLS * 4) // 262104

// ---------------------------------------------------------------------------
// Phase 1: norm2[l] = sum_c (mean of 2x2 block)^2, via V_WMMA_F32_16X16X4_F32.
// A = all-ones 16x4, so D[m][n] = sum_k B[k][n]: a per-column sum over the 4
// channels fed into B's four (VGPR, half-wave) slots. With A==1 the result is
// invariant to the K-slot permutation; only N = lane%16 matters (documented).
// Each wave: 16 columns, 64 iterations x 4 channels = 256 channels.
// ---------------------------------------------------------------------------
__global__ __launch_bounds__(256) void k_norm2(const float* __restrict__ x,
                                               float* __restrict__ norm2) {
  const int r     = blockIdx.x >> 1;     // pooled row 0..255
  const int half  = blockIdx.x & 1;
  const int wave  = threadIdx.x >> 5;
  const int lane  = threadIdx.x & 31;
  const int sbase = half * 128 + wave * 16;
  const int scol  = sbase + (lane & 15); // pooled col 0..255
  const int hi    = lane >> 4;           // half-wave selector

  const v2f ones = {1.0f, 1.0f};
  v8f acc = {};

  for (int it = 0; it < 64; ++it) {
    const int c0  = it * 4;
    const int ch0 = c0 + hi;       // B slot 0 channel
    const int ch1 = c0 + 2 + hi;   // B slot 1 channel

    const float2* p0 = (const float2*)(x + ((size_t)ch0 * H_IMG + 2 * r) * W_IMG + 2 * scol);
    const float2* p1 = (const float2*)(x + ((size_t)ch1 * H_IMG + 2 * r) * W_IMG + 2 * scol);
    float2 a0 = p0[0];
    float2 a1 = p0[W_IMG / 2];
    float2 b0 = p1[0];
    float2 b1 = p1[W_IMG / 2];

    float m0 = 0.25f * ((a0.x + a0.y) + (a1.x + a1.y));
    float m1 = 0.25f * ((b0.x + b0.y) + (b1.x + b1.y));
    v2f B = {m0 * m0, m1 * m1};

    acc = __builtin_amdgcn_wmma_f32_16x16x4_f32(false, ones, false, B,
                                                (short)0, acc, false, false);
  }
  // D layout: VGPR0, lanes 0..15 -> (M=0, N=lane); all rows equal (A==1).
  if (lane < 16) norm2[r * FW + sbase + lane] = acc[0];
}

// ---------------------------------------------------------------------------
// Phase 2: reproduce jax.random.uniform(key(42), (20,4)) via threefry-2x32,
// then sample_ind = argsort(u, axis=1)[:, :2].T  -> sample[n*20 + j].
// ---------------------------------------------------------------------------
__global__ void k_sample(int* __restrict__ sample) {
  if (threadIdx.x != 0 || blockIdx.x != 0) return;
  const unsigned ks0 = 0u, ks1 = 42u;
  const unsigned ks2 = ks0 ^ ks1 ^ 0x1BD11BDAu;
  float u[80];
  for (int f = 0; f < 40; ++f) {
    unsigned x0 = (unsigned)f + ks0;
    unsigned x1 = (unsigned)(f + 40) + ks1;
    const int rotA[4] = {13, 15, 26, 6};
    const int rotB[4] = {17, 29, 16, 24};
    for (int g = 0; g < 5; ++g) {
      const int* rot = (g & 1) ? rotB : rotA;
      for (int q = 0; q < 4; ++q) {
        x0 += x1;
        x1 = (x1 << rot[q]) | (x1 >> (32 - rot[q]));
        x1 ^= x0;
      }
      switch (g) {
        case 0: x0 += ks1; x1 += ks2 + 1u; break;
        case 1: x0 += ks2; x1 += ks0 + 2u; break;
        case 2: x0 += ks0; x1 += ks1 + 3u; break;
        case 3: x0 += ks1; x1 += ks2 + 4u; break;
        case 4: x0 += ks2; x1 += ks0 + 5u; break;
      }
    }
    u[f]      = __uint_as_float((x0 >> 9) | 0x3f800000u) - 1.0f;
    u[f + 40] = __uint_as_float((x1 >> 9) | 0x3f800000u) - 1.0f;
  }
  for (int i = 0; i < 20; ++i) {
    int used0 = -1;
    for (int n = 0; n < 2; ++n) {
      float best = 3.0f;
      int bidx = 0;
      for (int j = 0; j < 4; ++j) {
        if (j == used0) continue;
        float v = u[i * 4 + j];
        if (v < best) { best = v; bidx = j; }  // strict <  => stable tie-break
      }
      sample[n * 20 + i] = bidx;
      if (n == 0) used0 = bidx;
    }
  }
}

// ---------------------------------------------------------------------------
// Phase 3: exact stable rank of every column; order[rank] = column index.
// Norms (256KB) staged into LDS with global_load_async_to_lds_b128.
// ---------------------------------------------------------------------------
#define RANK_BLOCKS 128
#define IPB (LCOLS / RANK_BLOCKS) // 512 i's per block, 2 per thread

__global__ __launch_bounds__(256) void k_rank(const float* __restrict__ norm2,
                                              int* __restrict__ order) {
  extern __shared__ float snorm[]; // 65536 floats (256 KB dynamic LDS)
  const int tid = threadIdx.x;

  // Async fill: each thread issues 64 x 16B transfers (GVS: saddr + 32b voff).
  const unsigned ldsbase = __builtin_amdgcn_groupstaticsize();
  for (int k = 0; k < 64; ++k) {
    unsigned off      = (unsigned)((k * 256 + tid) * 16);
    unsigned lds_addr = ldsbase + off;
    asm volatile("global_load_async_to_lds_b128 %0, %1, %2"
                 :: "v"(lds_addr), "v"(off), "s"(norm2) : "memory");
  }
  asm volatile("s_wait_asynccnt 0" ::: "memory");
  __syncthreads();

  const int i0 = blockIdx.x * IPB + tid * 2;
  const int i1 = i0 + 1;
  const float k0 = snorm[i0];
  const float k1 = snorm[i1];
  int c0 = 0, c1 = 0;
  const float4* s4 = (const float4*)snorm;
  for (int jv = 0; jv < LCOLS / 4; ++jv) {
    float4 nj = s4[jv]; // uniform address across wave -> LDS broadcast
    int j = jv * 4;
    c0 += (nj.x < k0) || (nj.x == k0 && (j + 0) < i0);
    c0 += (nj.y < k0) || (nj.y == k0 && (j + 1) < i0);
    c0 += (nj.z < k0) || (nj.z == k0 && (j + 2) < i0);
    c0 += (nj.w < k0) || (nj.w == k0 && (j + 3) < i0);
    c1 += (nj.x < k1) || (nj.x == k1 && (j + 0) < i1);
    c1 += (nj.y < k1) || (nj.y == k1 && (j + 1) < i1);
    c1 += (nj.z < k1) || (nj.z == k1 && (j + 2) < i1);
    c1 += (nj.w < k1) || (nj.w == k1 && (j + 3) < i1);
  }
  order[c0] = i0;
  order[c1] = i1;
}

// ---------------------------------------------------------------------------
// Phase 4: t2 = all 4 patch elements of ranks 20..65535, rank-ordered.
// Coalesced float4 writes; gathered float2-pair reads (L2-resident per c).
// ---------------------------------------------------------------------------
__global__ __launch_bounds__(256) void k_t2(const float* __restrict__ x,
                                            const int* __restrict__ order,
                                            float* __restrict__ out) {
  const int rr = blockIdx.x * 256 + threadIdx.x; // rank - 20
  if (rr >= T2COLS) return;
  const int c    = blockIdx.y;
  const int col  = order[rr + KEEP];
  const int prow = col >> 8;
  const int pcol = col & 255;
  const float* p = x + ((size_t)c * H_IMG + prow * 2) * W_IMG + pcol * 2;
  float2 t = *(const float2*)p;
  float2 b = *(const float2*)(p + W_IMG);
  *(float4*)(out + (size_t)c * OUTC + KEEP * 2 + (size_t)rr * 4) =
      make_float4(t.x, t.y, b.x, b.y); // k = kh*2 + kw order
}

// ---------------------------------------------------------------------------
// Phase 5: t1 = 2 sampled patch elements of the 20 smallest-norm columns.
// ---------------------------------------------------------------------------
__global__ void k_t1(const float* __restrict__ x, const int* __restrict__ order,
                     const int* __restrict__ sample, float* __restrict__ out) {
  const int t = blockIdx.x * 256 + threadIdx.x;
  if (t >= C_CH * KEEP * 2) return;
  const int c   = t / 40;
  const int rem = t % 40;
  const int j   = rem >> 1;
  const int n   = rem & 1;
  const int col = order[j];
  const int k   = sample[n * 20 + j];
  const int kh = k >> 1, kw = k & 1;
  const int prow = col >> 8, pcol = col & 255;
  out[(size_t)c * OUTC + j * 2 + n] =
      x[((size_t)c * H_IMG + prow * 2 + kh) * W_IMG + pcol * 2 + kw];
}

extern "C" void kernel_launch(void* const* d_in, const int* in_sizes, int n_in,
                              void* d_out, int out_size, void* d_ws, size_t ws_size,
                              hipStream_t stream) {
  (void)in_sizes; (void)n_in; (void)out_size; (void)ws_size;
  const float* x = (const float*)d_in[0];
  float* out = (float*)d_out;

  float* norm2 = (float*)d_ws;                                   // 256 KB
  int*   order = (int*)((char*)d_ws + LCOLS * sizeof(float));    // 256 KB
  int*   samp  = (int*)((char*)d_ws + 2 * LCOLS * sizeof(float)); // 160 B

  k_norm2<<<512, 256, 0, stream>>>(x, norm2);
  k_sample<<<1, 32, 0, stream>>>(samp);
  k_rank<<<RANK_BLOCKS, 256, LCOLS * sizeof(float), stream>>>(norm2, order);
  dim3 g2((T2COLS + 255) / 256, C_CH);
  k_t2<<<g2, 256, 0, stream>>>(x, order, out);
  k_t1<<<(C_CH * KEEP * 2 + 255) / 256, 256, 0, stream>>>(x, order, samp, out);
}